// DCN_2628519985380
// MI455X (gfx1250) — compile-verified
//
#include <hip/hip_runtime.h>
#include <hip/hip_bf16.h>

// ---------------------------------------------------------------------------
// DCNv2 fused: offset-conv -> bilinear setup -> LDS-staged WMMA f32 GEMM
// Shapes: B=4, C=64, H=W=128, O=64, KH=KW=3 (K=9), stride=1, pad=1, dil=1
// GEMM view: out[o, pix] = sum_{ck} W[o,ck] * val[ck,pix],
//            M=O=64, N=B*HW=65536, K=C*K=576 (ordered ck = k*64 + c).
// ---------------------------------------------------------------------------

#define B_   4
#define C_   64
#define H_   128
#define W_   128
#define O_   64
#define K_   9
#define HW_  (H_ * W_)          // 16384
#define CK_  (C_ * K_)          // 576
#define NPIX (B_ * HW_)         // 65536
#define WLDS_F2 (CK_ / 2 * O_)  // 18432 float2 = 147456 bytes

typedef __attribute__((ext_vector_type(2))) float v2f;
typedef __attribute__((ext_vector_type(8))) float v8f;

// ---------------------------------------------------------------------------
// 1) NCHW -> NHWC transpose of x (channel pairs contiguous for v2f gathers)
// ---------------------------------------------------------------------------
__global__ void __launch_bounds__(256) nhwc_kernel(const float* __restrict__ x,
                                                   float* __restrict__ xT) {
    int gid = blockIdx.x * 256 + threadIdx.x;      // NHWC linear index
    if (gid >= B_ * HW_ * C_) return;
    int c  = gid & (C_ - 1);
    int hw = (gid >> 6) & (HW_ - 1);
    int b  = gid >> 20;                            // 16384*64 = 2^20
    xT[gid] = x[(((b << 6) + c) << 14) + hw];
}

// ---------------------------------------------------------------------------
// 2) Weight repack: weight[O][C][K] -> wT2[j][o] = (w[o][ck=2j], w[o][ck=2j+1])
//    K-dim ordering ck = k*C_ + c (k outer: coords fixed across 64 channels)
// ---------------------------------------------------------------------------
__global__ void __launch_bounds__(256) wprep_kernel(const float* __restrict__ w,
                                                    float2* __restrict__ wT2) {
    int gid = blockIdx.x * 256 + threadIdx.x;      // 288*64 = 18432
    if (gid >= WLDS_F2) return;
    int o = gid & (O_ - 1);
    int j = gid >> 6;                              // 0..287
    int q0 = 2 * j, q1 = 2 * j + 1;
    int k0 = q0 >> 6, c0 = q0 & 63;
    int k1 = q1 >> 6, c1 = q1 & 63;
    float2 r;
    r.x = w[(o * C_ + c0) * K_ + k0];
    r.y = w[(o * C_ + c1) * K_ + k1];
    wT2[gid] = r;
}

// ---------------------------------------------------------------------------
// 3) Offset conv (27ch 3x3) + sigmoid + bilinear setup.
//    One thread per (b,k,pixel). Stores:
//      samp: float4 corner weights, premultiplied by mask and validity
//      soff: int4  clamped NHWC element offsets ((y<<7 | x) << 6), per corner
// ---------------------------------------------------------------------------
__global__ void __launch_bounds__(256) offset_kernel(const float* __restrict__ x,
                                                     const float* __restrict__ cow,
                                                     const float* __restrict__ cob,
                                                     float4* __restrict__ samp,
                                                     int4*   __restrict__ soff) {
    int gid = blockIdx.x * 256 + threadIdx.x;      // B_*K_*HW_ = 589824
    if (gid >= B_ * K_ * HW_) return;
    int hw = gid & (HW_ - 1);
    int bk = gid >> 14;
    int k  = bk % K_;
    int b  = bk / K_;
    int ho = hw >> 7;
    int wo = hw & (W_ - 1);

    const int chy = 2 * k, chx = 2 * k + 1, chm = 2 * K_ + k;
    float sdy = cob[chy], sdx = cob[chx], sm = cob[chm];

    const float* xb = x + ((size_t)b * C_) * HW_;
    for (int c = 0; c < C_; ++c) {
        const float* xc = xb + (size_t)c * HW_;
        const float* wy = cow + ((size_t)chy * C_ + c) * K_;
        const float* wx = cow + ((size_t)chx * C_ + c) * K_;
        const float* wm = cow + ((size_t)chm * C_ + c) * K_;
        #pragma unroll
        for (int kh = 0; kh < 3; ++kh) {
            int yy = ho - 1 + kh;
            if (yy < 0 || yy >= H_) continue;
            #pragma unroll
            for (int kw = 0; kw < 3; ++kw) {
                int xx = wo - 1 + kw;
                if (xx < 0 || xx >= W_) continue;
                float v = xc[(yy << 7) + xx];
                int kk = kh * 3 + kw;
                sdy = fmaf(v, wy[kk], sdy);
                sdx = fmaf(v, wx[kk], sdx);
                sm  = fmaf(v, wm[kk], sm);
            }
        }
    }
    float mask = 1.0f / (1.0f + __expf(-sm));

    int ki = k / 3, kj = k % 3;
    float py = sdy + (float)(ho - 1 + ki);
    float px = sdx + (float)(wo - 1 + kj);
    float y0f = floorf(py), x0f = floorf(px);
    float ty = py - y0f, tx = px - x0f;
    int y0 = (int)y0f, x0 = (int)x0f;
    int y1 = y0 + 1,   x1 = x0 + 1;

    bool vy0 = (y0 >= 0) & (y0 < H_);
    bool vy1 = (y1 >= 0) & (y1 < H_);
    bool vx0 = (x0 >= 0) & (x0 < W_);
    bool vx1 = (x1 >= 0) & (x1 < W_);

    float4 wgt;
    wgt.x = (vy0 && vx0) ? (1.0f - ty) * (1.0f - tx) * mask : 0.0f;
    wgt.y = (vy0 && vx1) ? (1.0f - ty) * tx          * mask : 0.0f;
    wgt.z = (vy1 && vx0) ? ty          * (1.0f - tx) * mask : 0.0f;
    wgt.w = (vy1 && vx1) ? ty          * tx          * mask : 0.0f;

    int y0c = min(max(y0, 0), H_ - 1);
    int y1c = min(max(y1, 0), H_ - 1);
    int x0c = min(max(x0, 0), W_ - 1);
    int x1c = min(max(x1, 0), W_ - 1);

    int4 of;
    of.x = (((y0c << 7) + x0c) << 6);
    of.y = (((y0c << 7) + x1c) << 6);
    of.z = (((y1c << 7) + x0c) << 6);
    of.w = (((y1c << 7) + x1c) << 6);

    samp[gid] = wgt;
    soff[gid] = of;
}

// ---------------------------------------------------------------------------
// 4) Fused gather + WMMA f32 GEMM, weights staged in LDS.
//    512 threads = 16 waves; each wave: 16 pixels x full M=64 (4 accumulators).
//    Outer loop over k (9): one samp/soff load; inner 16 steps of K+=4 use
//    immediate-offset gathers from 4 corner base pointers; B-frag built with
//    v_pk_mul/v_pk_fma on v2f vectors.
//    B-frag: lane l -> N = l&15; K-slots (l<16 ? {+0,+1} : {+2,+3}).
//    A-frag: lane l -> M = l&15; same K-slot split; read from LDS as v2f.
// ---------------------------------------------------------------------------
__global__ void __launch_bounds__(512) dcn_gemm_kernel(const float*  __restrict__ xT,
                                                       const float4* __restrict__ samp,
                                                       const int4*   __restrict__ soff,
                                                       const float2* __restrict__ wT2,
                                                       const float*  __restrict__ bias,
                                                       float* __restrict__ out) {
    extern __shared__ v2f wlds[];                  // 18432 v2f = 147456 B

    // --- cooperative stage of repacked weights into LDS (float4 moves) ---
    {
        const float4* src = (const float4*)wT2;
        float4* dst = (float4*)wlds;
        #pragma unroll
        for (int i = 0; i < WLDS_F2 / 2 / 512; ++i)          // 18 iters
            dst[i * 512 + threadIdx.x] = src[i * 512 + threadIdx.x];
    }
    __syncthreads();

    const int lane   = threadIdx.x & 31;
    const int wave   = threadIdx.x >> 5;           // 0..15
    const int lane16 = lane & 15;
    const int hi     = lane >> 4;                  // 0 / 1
    const int pixel  = blockIdx.x * 256 + wave * 16 + lane16; // < 65536
    const int b      = pixel >> 14;
    const int hw     = pixel & (HW_ - 1);

    // this lane's channel-pair base offset within a 64-channel pixel record
    const float* xb = xT + ((size_t)b * HW_) * C_ + (hi << 1);

    v8f acc[4];
    #pragma unroll
    for (int m = 0; m < 4; ++m) acc[m] = (v8f){0.f,0.f,0.f,0.f,0.f,0.f,0.f,0.f};

    for (int k = 0; k < K_; ++k) {
        const int s = ((b * K_ + k) << 14) + hw;
        const float4 wgt = samp[s];
        const int4   of  = soff[s];

        const float* p00 = xb + of.x;
        const float* p01 = xb + of.y;
        const float* p10 = xb + of.z;
        const float* p11 = xb + of.w;

        // LDS row base for this k and this lane's K-slot half
        const v2f* wrow = wlds + (((k << 5) + hi) << 6) + lane16;

        #pragma unroll
        for (int i = 0; i < 16; ++i) {             // K-steps of 4 within this k
            const int co = i << 2;                 // channel advance (elements)
            const v2f v00 = *(const v2f*)(p00 + co);
            const v2f v01 = *(const v2f*)(p01 + co);
            const v2f v10 = *(const v2f*)(p10 + co);
            const v2f v11 = *(const v2f*)(p11 + co);

            // 1x v_pk_mul_f32 + 3x v_pk_fma_f32
            v2f bfr = v00 * wgt.x;
            bfr = __builtin_elementwise_fma((v2f)(wgt.y), v01, bfr);
            bfr = __builtin_elementwise_fma((v2f)(wgt.z), v10, bfr);
            bfr = __builtin_elementwise_fma((v2f)(wgt.w), v11, bfr);

            // A fragments from LDS: row j = k*32 + i*2 + hi, entry mt*16+lane16
            const v2f* wr = wrow + (i << 7);       // i*2 rows of 64 v2f
            const v2f af0 = wr[0];
            const v2f af1 = wr[16];
            const v2f af2 = wr[32];
            const v2f af3 = wr[48];

            acc[0] = __builtin_amdgcn_wmma_f32_16x16x4_f32(false, af0, false, bfr,
                                                           (short)0, acc[0], false, false);
            acc[1] = __builtin_amdgcn_wmma_f32_16x16x4_f32(false, af1, false, bfr,
                                                           (short)0, acc[1], false, false);
            acc[2] = __builtin_amdgcn_wmma_f32_16x16x4_f32(false, af2, false, bfr,
                                                           (short)0, acc[2], false, false);
            acc[3] = __builtin_amdgcn_wmma_f32_16x16x4_f32(false, af3, false, bfr,
                                                           (short)0, acc[3], false, false);
        }
    }

    // Epilogue: D layout — VGPR v of acc[mt] holds M = mt*16 + v + 8*hi, N = lane16.
    float* outp = out + ((size_t)b * O_) * HW_ + hw;
    #pragma unroll
    for (int mt = 0; mt < 4; ++mt) {
        #pragma unroll
        for (int v = 0; v < 8; ++v) {
            int o = mt * 16 + v + (hi << 3);
            outp[(size_t)o * HW_] = acc[mt][v] + bias[o];
        }
    }
}

// ---------------------------------------------------------------------------
// Launch
// ---------------------------------------------------------------------------
extern "C" void kernel_launch(void* const* d_in, const int* in_sizes, int n_in,
                              void* d_out, int out_size, void* d_ws, size_t ws_size,
                              hipStream_t stream) {
    const float* x      = (const float*)d_in[0];   // (4,64,128,128)
    const float* cow    = (const float*)d_in[1];   // (27,64,3,3)
    const float* cob    = (const float*)d_in[2];   // (27,)
    const float* weight = (const float*)d_in[3];   // (64,64,3,3)
    const float* bias   = (const float*)d_in[4];   // (64,)
    float* out = (float*)d_out;                    // (4,64,128,128)

    // workspace layout
    float*  xT   = (float*)d_ws;                          // 4,194,304 f  (16 MB)
    float4* samp = (float4*)(xT + (size_t)B_ * HW_ * C_); // 589,824 f4   (9.4 MB)
    int4*   soff = (int4*)(samp + (size_t)B_ * K_ * HW_); // 589,824 i4   (9.4 MB)
    float2* wT2  = (float2*)(soff + (size_t)B_ * K_ * HW_); // 18,432 f2  (147 KB)

    {   // 1) NHWC transpose
        int n = B_ * HW_ * C_;
        nhwc_kernel<<<(n + 255) / 256, 256, 0, stream>>>(x, xT);
    }
    {   // 2) weight repack
        wprep_kernel<<<(WLDS_F2 + 255) / 256, 256, 0, stream>>>(weight, wT2);
    }
    {   // 3) offset conv + bilinear setup
        int n = B_ * K_ * HW_;
        offset_kernel<<<(n + 255) / 256, 256, 0, stream>>>(x, cow, cob, samp, soff);
    }
    {   // 4) fused gather + WMMA GEMM: 512 threads, 256 pixels/block,
        //    147456 B dynamic LDS holding the full repacked weight
        dcn_gemm_kernel<<<NPIX / 256, 512, (size_t)WLDS_F2 * sizeof(float2), stream>>>(
            xT, samp, soff, wT2, bias, out);
    }
}